// AFTLocal_87170656240174
// MI455X (gfx1250) — compile-verified
//
#include <hip/hip_runtime.h>

typedef __attribute__((ext_vector_type(2))) float v2f;
typedef __attribute__((ext_vector_type(8))) float v8f;

#define B_  4
#define L_  1024
#define E_  256
#define S_  32
#define W_  (2 * S_ + 1)
#define M_  (B_ * L_)          // 4096 rows
#define NEGINF (-1e30f)

// ---------------------------------------------------------------------------
// Projection GEMM: out[m,n] = sum_k x[m,k] * Wt[n,k] + bias[n]
// One wave computes a 16(M) x 64(N) strip = 4 output tiles via
// V_WMMA_F32_16X16X4_F32, reusing the A fragment across the 4 N-tiles
// (1.25 fragment loads per WMMA) with 4 independent accumulator chains.
//
// Fragment layouts (CDNA5 ISA 7.12.2, wave32):
//  A 16x4 f32 : lane<16 -> row M=lane, {K=0,K=1}; lane>=16 -> row M=lane-16, {K=2,K=3}
//  B 4x16 f32 : lane<16 -> col N=lane, rows {0,1}; lane>=16 -> col N=lane-16, rows {2,3}
//  C/D 16x16  : VGPR v, lane<16 -> M=v, N=lane; lane>=16 -> M=v+8, N=lane-16
// ---------------------------------------------------------------------------
__global__ __launch_bounds__(256) void proj_wmma_kernel(
    const float* __restrict__ x,     // [M_, E_]
    const float* __restrict__ Wt,    // [E_, E_]  (row n holds weights of output n)
    const float* __restrict__ bias,  // [E_]
    float* __restrict__ out)         // [M_, E_]
{
    const int lane = threadIdx.x & 31;
    const int wave = threadIdx.x >> 5;
    const int job  = blockIdx.x * 8 + wave;      // 8 waves / block, 1024 jobs
    const int tm   = job >> 2;                   // 0..255  (M_/16 m-tiles)
    const int tn   = job & 3;                    // 0..3    (E_/64 n-strips)
    const int m0   = tm * 16;
    const int n0   = tn * 64;
    const int half = lane >> 4;                  // 0 or 1
    const int lid  = lane & 15;

    const float* arow = x  + (size_t)(m0 + lid) * E_ + 2 * half;
    const float* brow = Wt + (size_t)(n0 + lid) * E_ + 2 * half;

    v8f c0 = {0.f, 0.f, 0.f, 0.f, 0.f, 0.f, 0.f, 0.f};
    v8f c1 = c0, c2 = c0, c3 = c0;

#pragma unroll 4
    for (int k = 0; k < E_; k += 4) {
        v2f a  = *(const v2f*)(arow + k);
        v2f b0 = *(const v2f*)(brow + k);
        v2f b1 = *(const v2f*)(brow + 16 * E_ + k);
        v2f b2 = *(const v2f*)(brow + 32 * E_ + k);
        v2f b3 = *(const v2f*)(brow + 48 * E_ + k);
        // (neg_a, A, neg_b, B, c_mod, C, reuse_a, reuse_b)
        c0 = __builtin_amdgcn_wmma_f32_16x16x4_f32(false, a, false, b0, (short)0, c0, false, false);
        c1 = __builtin_amdgcn_wmma_f32_16x16x4_f32(false, a, false, b1, (short)0, c1, false, false);
        c2 = __builtin_amdgcn_wmma_f32_16x16x4_f32(false, a, false, b2, (short)0, c2, false, false);
        c3 = __builtin_amdgcn_wmma_f32_16x16x4_f32(false, a, false, b3, (short)0, c3, false, false);
    }

    const float bn0 = bias[n0 + lid];
    const float bn1 = bias[n0 + 16 + lid];
    const float bn2 = bias[n0 + 32 + lid];
    const float bn3 = bias[n0 + 48 + lid];

    float* orow = out + (size_t)(m0 + 8 * half) * E_ + n0 + lid;
#pragma unroll
    for (int v = 0; v < 8; ++v) {
        float* p = orow + (size_t)v * E_;
        p[0]  = c0[v] + bn0;
        p[16] = c1[v] + bn1;
        p[32] = c2[v] + bn2;
        p[48] = c3[v] + bn3;
    }
}

// ---------------------------------------------------------------------------
// Fused AFT-local window stage. Block = one (b,l) row, thread = channel e.
// Per-channel online softmax over the W_=65 window:
//   strict (|rel|<S): K_win = K + pos_bias, else K_win = 0 (logit 0, kept)
//   valid  (0<=j<L):  else logit = -1e30 (excluded) and V contribution = 0
// Final: sigmoid(Q)^2 * (sum w*V / sum w)
// ---------------------------------------------------------------------------
__global__ __launch_bounds__(256) void aft_local_kernel(
    const float* __restrict__ Q,
    const float* __restrict__ K,
    const float* __restrict__ V,
    const float* __restrict__ pb,    // [W_, E_]
    float* __restrict__ out)         // [B_, L_, E_]
{
    const int e     = threadIdx.x;       // 0..255
    const int bl    = blockIdx.x;        // 0..4095 (flattened b*L_ + l)
    const int l     = bl & (L_ - 1);
    const int bbase = bl - l;            // b * L_
    const size_t rowbase = (size_t)bl * E_;

    const float qv = Q[rowbase + e];

    float m   = -3.0e38f;   // running max
    float s   = 0.f;        // running denom
    float acc = 0.f;        // running numer (weighted V)

#pragma unroll 1
    for (int w = 0; w < W_; ++w) {
        const int rel = w - S_;
        const int j   = l + rel;
        const bool valid  = (j >= 0) & (j < L_);
        const bool strict = (rel > -S_) & (rel < S_);
        const int jc = valid ? j : 0;
        const size_t rb = (size_t)(bbase + jc) * E_ + e;

        const float kv = strict ? (K[rb] + pb[w * E_ + e]) : 0.f;
        const float g  = valid ? (qv * kv) : NEGINF;
        const float vv = valid ? V[rb] : 0.f;

        const float nm = fmaxf(m, g);
        const float sc = __expf(m - nm);   // underflows to 0 for the -inf start
        const float p  = __expf(g - nm);
        s   = s * sc + p;
        acc = acc * sc + p * vv;
        m = nm;
    }

    const float ctx = acc / s;             // s > 0: rel==0 is always valid
    const float sig = 1.f / (1.f + __expf(-qv));
    out[rowbase + e] = sig * sig * ctx;
}

// ---------------------------------------------------------------------------
extern "C" void kernel_launch(void* const* d_in, const int* in_sizes, int n_in,
                              void* d_out, int out_size, void* d_ws, size_t ws_size,
                              hipStream_t stream)
{
    const float* q  = (const float*)d_in[0];
    const float* Wq = (const float*)d_in[1];
    const float* bq = (const float*)d_in[2];
    const float* Wk = (const float*)d_in[3];
    const float* bk = (const float*)d_in[4];
    const float* Wv = (const float*)d_in[5];
    const float* bv = (const float*)d_in[6];
    const float* pb = (const float*)d_in[7];
    float* out = (float*)d_out;

    float* ws = (float*)d_ws;
    const size_t plane = (size_t)M_ * E_;   // 1,048,576 floats = 4 MB
    float* Qp = ws;
    float* Kp = ws + plane;
    float* Vp = ws + 2 * plane;

    const dim3 block(256);
    const dim3 grid_gemm((M_ / 16) * (E_ / 64) / 8);  // 128 blocks, 8 wave-jobs each

    proj_wmma_kernel<<<grid_gemm, block, 0, stream>>>(q, Wq, bq, Qp);
    proj_wmma_kernel<<<grid_gemm, block, 0, stream>>>(q, Wk, bk, Kp);
    proj_wmma_kernel<<<grid_gemm, block, 0, stream>>>(q, Wv, bv, Vp);

    aft_local_kernel<<<dim3(M_), block, 0, stream>>>(Qp, Kp, Vp, pb, out);
}